// GCDDLayer_11158325035386
// MI455X (gfx1250) — compile-verified
//
#include <hip/hip_runtime.h>
#include <cstdint>

// GCDD layer fused kernel for MI455X (gfx1250, wave32).
// One pass: u -> (ux,uy) -> (uxx,uxy,uyy,G,phi,P,Q) -> u + div(P,Q), all in LDS.
// Global traffic = read u once (async global->LDS b128 DMA, ASYNCcnt) + write
// out once (b128 stores) -- the HBM roofline minimum (~96 MiB @ 23.3 TB/s).
// WMMA is deliberately not used: a 3x3 depthwise stencil as a banded matmul
// has <10% useful MACs per v_wmma issue slot, worse than v_fma_f32.

#define TILE 32            // output tile edge
#define R0H  (TILE + 6)    // u rows   (halo 3 top/bottom)            = 38
#define SUW  (TILE + 8)    // u stride (halo 4 left, 3+1 pad right)   = 40
#define R1   (TILE + 4)    // ux/uy valid region (halo 2)             = 36
#define SXW  40            // ux/uy stride (padded, 16B-aligned rows)
#define R2   (TILE + 2)    // P/Q valid region (halo 1)               = 34
#define SPW  36            // P/Q stride (34 valid + 2 junk cols)
#define IMG_H 512
#define IMG_W 512
#define NTHREADS 256       // 8 wave32 waves

__global__ __launch_bounds__(NTHREADS)
void gcdd_fused_kernel(const float* __restrict__ u, float* __restrict__ out) {
    __shared__ __align__(16) float sU [R0H * SUW];   // 38 x 40
    __shared__ __align__(16) float sUX[R1  * SXW];   // 36 x 40 (36 valid cols)
    __shared__ __align__(16) float sUY[R1  * SXW];
    __shared__ __align__(16) float sP [R2  * SPW];   // 34 x 36 (34 valid cols)
    __shared__ __align__(16) float sQ [R2  * SPW];

    const int tid   = threadIdx.x;
    const int x0    = blockIdx.x * TILE;
    const int y0    = blockIdx.y * TILE;
    const int plane = blockIdx.z;

    const float* __restrict__ up = u   + (size_t)plane * IMG_H * IMG_W;
    float*       __restrict__ op = out + (size_t)plane * IMG_H * IMG_W;

    // Low 32 bits of a flat pointer to a __shared__ object = LDS byte address
    // (aperture is in the high 32 bits) -- what the async-to-LDS VDST expects.
    const unsigned sU_lds = (unsigned)(size_t)(&sU[0]);

    // ---- Stage 0: u tile (38 rows x 40 cols, halo) -> LDS via async b128 ---
    // u grid (r,c) covers global (y0-3+r, x0-4+c). Column blocks of 4 floats
    // are 16B aligned in global AND LDS, and each block is entirely inside or
    // entirely outside the image (x0 % 32 == 0, 512 % 4 == 0).
    for (int i = tid; i < R0H * (SUW / 4); i += NTHREADS) {   // 380 blocks
        const int r   = i / (SUW / 4);
        const int cb  = i - r * (SUW / 4);
        const int gy  = y0 - 3 + r;
        const int gxb = x0 - 4 + 4 * cb;
        if ((unsigned)gy < IMG_H && (unsigned)gxb < IMG_W) {
            const unsigned lds_off = sU_lds + 4u * (unsigned)(r * SUW + 4 * cb);
            const int      goff    = (gy * IMG_W + gxb) * 4;
            asm volatile("global_load_async_to_lds_b128 %0, %1, %2"
                         :
                         : "v"(lds_off), "v"(goff), "s"(up)
                         : "memory");
        } else {
            *(float4*)&sU[r * SUW + 4 * cb] = make_float4(0.f, 0.f, 0.f, 0.f);
        }
    }
#if __has_builtin(__builtin_amdgcn_s_wait_asynccnt)
    __builtin_amdgcn_s_wait_asynccnt(0);
#else
    asm volatile("s_wait_asynccnt 0x0" ::: "memory");
#endif
    __syncthreads();

    // ---- Stage 1: first derivatives on 36x36, 4 columns per task ----------
    // ux point (r,c) -> global (y0-2+r, x0-2+c): u rows r..r+2, u cols c+1..c+3.
    for (int i = tid; i < R1 * 9; i += NTHREADS) {            // 324 tasks
        const int r  = i / 9;
        const int c0 = (i - r * 9) * 4;                       // 0..32
        const float* U0 = &sU[r * SUW + (c0 + 1)];
        const float* U1 = U0 + SUW;
        const float* U2 = U1 + SUW;
        float a0[6], a1[6], a2[6];
#pragma unroll
        for (int j = 0; j < 6; ++j) { a0[j] = U0[j]; a1[j] = U1[j]; a2[j] = U2[j]; }
        float4 vx, vy;
        float* px = &vx.x;
        float* py = &vy.x;
#pragma unroll
        for (int j = 0; j < 4; ++j) {
            px[j] = (a0[j + 2] - a0[j]) + 2.0f * (a1[j + 2] - a1[j]) + (a2[j + 2] - a2[j]);
            py[j] = (a2[j] - a0[j]) + 2.0f * (a2[j + 1] - a0[j + 1]) + (a2[j + 2] - a0[j + 2]);
        }
        *(float4*)&sUX[r * SXW + c0] = vx;
        *(float4*)&sUY[r * SXW + c0] = vy;
    }
    __syncthreads();

    // ---- Stage 2: 2nd derivatives, curvature, P/Q; 4 columns per task -----
    // P point (r,c) -> uses UX/UY rows r..r+2, cols c..c+2. Columns 34..35 are
    // junk (read padded LDS, never consumed downstream).
    for (int i = tid; i < R2 * 9; i += NTHREADS) {            // 306 tasks
        const int r  = i / 9;
        const int c0 = (i - r * 9) * 4;
        const float* X0 = &sUX[r * SXW + c0];
        const float* X1 = X0 + SXW;
        const float* X2 = X1 + SXW;
        const float* Y0 = &sUY[r * SXW + c0];
        const float* Y1 = Y0 + SXW;
        const float* Y2 = Y1 + SXW;
        float x0r[6], x1r[6], x2r[6], y0r[6], y1r[6], y2r[6];
#pragma unroll
        for (int j = 0; j < 6; ++j) {
            x0r[j] = X0[j]; x1r[j] = X1[j]; x2r[j] = X2[j];
            y0r[j] = Y0[j]; y1r[j] = Y1[j]; y2r[j] = Y2[j];
        }
        float4 vp, vq;
        float* pp = &vp.x;
        float* pq = &vq.x;
#pragma unroll
        for (int j = 0; j < 4; ++j) {
            const float uxx = (x0r[j + 2] - x0r[j]) + 2.0f * (x1r[j + 2] - x1r[j]) + (x2r[j + 2] - x2r[j]);
            const float uxy = (x2r[j] - x0r[j]) + 2.0f * (x2r[j + 1] - x0r[j + 1]) + (x2r[j + 2] - x0r[j + 2]);
            const float uyy = (y2r[j] - y0r[j]) + 2.0f * (y2r[j + 1] - y0r[j + 1]) + (y2r[j + 2] - y0r[j + 2]);
            const float xc  = x1r[j + 1];
            const float yc  = y1r[j + 1];
            const float nb    = 1.0f + xc * xc + yc * yc;
            const float denom = nb * nb + 1e-6f;
            const float G     = (uxx * uyy - uxy * uxy) / denom;
            const float phi   = __expf(-fabsf(G));            // v_exp_f32 path
            pp[j] = phi * xc;
            pq[j] = phi * yc;
        }
        *(float4*)&sP[r * SPW + c0] = vp;
        *(float4*)&sQ[r * SPW + c0] = vq;
    }
    __syncthreads();

    // ---- Stage 3: divergence + residual; one float4 store per thread ------
    // 32 rows x 8 col-blocks = 256 tasks, exactly one per thread.
    {
        const int r  = tid >> 3;
        const int c0 = (tid & 7) * 4;
        const float* P0 = &sP[r * SPW + c0];
        const float* P1 = P0 + SPW;
        const float* P2 = P1 + SPW;
        const float* Q0 = &sQ[r * SPW + c0];
        const float* Q2 = Q0 + 2 * SPW;
        float p0[6], p1[6], p2[6], q0[6], q2[6];
#pragma unroll
        for (int j = 0; j < 6; ++j) {
            p0[j] = P0[j]; p1[j] = P1[j]; p2[j] = P2[j];
            q0[j] = Q0[j]; q2[j] = Q2[j];
        }
        const float4 uc4 = *(const float4*)&sU[(r + 3) * SUW + (c0 + 4)];
        const float* uc = &uc4.x;
        float4 o4;
        float* o = &o4.x;
#pragma unroll
        for (int j = 0; j < 4; ++j) {
            const float divP = (p0[j + 2] - p0[j]) + 2.0f * (p1[j + 2] - p1[j]) + (p2[j + 2] - p2[j]);
            const float divQ = (q2[j] - q0[j]) + 2.0f * (q2[j + 1] - q0[j + 1]) + (q2[j + 2] - q0[j + 2]);
            o[j] = uc[j] + divP + divQ;
        }
        *(float4*)&op[(size_t)(y0 + r) * IMG_W + (x0 + c0)] = o4;
    }
}

extern "C" void kernel_launch(void* const* d_in, const int* in_sizes, int n_in,
                              void* d_out, int out_size, void* d_ws, size_t ws_size,
                              hipStream_t stream) {
    const float* u = (const float*)d_in[0];
    // d_in[1] is theta; the reference computes theta*u and discards it.
    float* out = (float*)d_out;

    const int planes = in_sizes[0] / (IMG_H * IMG_W);   // 16*3 = 48
    dim3 grid(IMG_W / TILE, IMG_H / TILE, planes);
    gcdd_fused_kernel<<<grid, NTHREADS, 0, stream>>>(u, out);
}